// SharedRoutingBatchedForest_37804302139408
// MI455X (gfx1250) — compile-verified
//
#include <hip/hip_runtime.h>
#include <hip/hip_bf16.h>

typedef float v2f __attribute__((ext_vector_type(2)));
typedef float v8f __attribute__((ext_vector_type(8)));
typedef unsigned int u32;
typedef u32 u32x4 __attribute__((ext_vector_type(4)));
typedef int  i32x4 __attribute__((ext_vector_type(4)));
typedef int  i32x8 __attribute__((ext_vector_type(8)));

#define D_IN   1024
#define D_OUT  1024
#define NTOK   8192     // B*S = 4*2048
#define NCOLS  96       // 84 decision logits + 12 gate logits
#define KC     32       // K chunk for kernel 1
#define XSTR   34       // kernel-1 LDS stride (even -> 8B-aligned float2, conflict-free)
#define LSTR   101      // logits LDS stride (odd -> conflict-free b32 row reads)
#define ASTR   102      // kernel-2 A-tile stride = 96 + 2 pad per 32 dwords (TDM pad_enable)
#define BSTR   98       // kernel-2 B-tile stride (even, conflict-free)

__device__ __forceinline__ float sigmoidf_(float x) { return 1.0f / (1.0f + expf(-x)); }

// ---------------------------------------------------------------------------
// Kernel 1: logits = x @ [decision_weights ; gate_w]^T  (M=8192, N=96, K=1024)
// 64-token tiles (128 WGs to fill the machine); waves split 4(M) x 2(N).
// Epilogue: sigmoid -> depth-3 path products -> tree softmax -> C[8192][96].
// ---------------------------------------------------------------------------
__global__ __launch_bounds__(256)
void forest_route_kernel(const float* __restrict__ x,    // [8192][1024]
                         const float* __restrict__ dw,   // [84][1024]
                         const float* __restrict__ db,   // [84]
                         const float* __restrict__ ntl,  // [84]
                         const float* __restrict__ gw,   // [12][1024]
                         const float* __restrict__ gb,   // [12]
                         float* __restrict__ C)          // [8192][96]
{
    // LDS: invT[84] | bo[84] | (Xs[64][34] + Ws[96][34]) reused as Ls[64][101]
    __shared__ float smem[168 + 64 * LSTR];
    float* invT = smem;
    float* bo   = smem + 84;
    float* Xs   = smem + 168;
    float* Ws   = smem + 168 + 64 * XSTR;
    float* Ls   = smem + 168;

    const int tid    = threadIdx.x;
    const int lane   = tid & 31;
    const int w      = tid >> 5;        // wave 0..7
    const int laneLo = lane & 15;
    const int laneHi = lane >> 4;       // 0/1 -> K offset +0/+2 in fragments
    const int waveM  = w & 3;           // 16-token strip
    const int nb     = (w >> 2) * 3;    // first 16-col block (0 or 3)
    const int mBase  = blockIdx.x * 64;

    if (tid < 84) {
        float z  = ntl[tid] + 0.5413f;
        float sp = (z > 20.0f) ? z : log1pf(expf(z));
        float it = 1.0f / sp;           // TEMPERATURE == 1
        invT[tid] = it;
        bo[tid]   = db[tid] * it;
    }

    v8f acc[3] = {};   // 16 tokens x 48 logit cols per wave

    for (int k0 = 0; k0 < D_IN; k0 += KC) {
        __syncthreads();
        // stage x tile [64][32]
        #pragma unroll
        for (int i = 0; i < 8; ++i) {
            int e = tid + i * 256;
            int tok = e >> 5, kk = e & 31;
            Xs[tok * XSTR + kk] = x[(size_t)(mBase + tok) * D_IN + k0 + kk];
        }
        // stage W tile [96][32]: rows 0..83 = decision weights, 84..95 = gate weights
        #pragma unroll
        for (int i = 0; i < 12; ++i) {
            int e = tid + i * 256;
            int row = e >> 5, kk = e & 31;
            Ws[row * XSTR + kk] = (row < 84)
                ? dw[(size_t)row * D_IN + k0 + kk]
                : gw[(size_t)(row - 84) * D_IN + k0 + kk];
        }
        if (k0 + KC < D_IN) {
            int e = tid << 3;
            int tok = e >> 5, kk = e & 31;
            __builtin_prefetch(&x[(size_t)(mBase + tok) * D_IN + k0 + KC + kk], 0, 1);
        }
        __syncthreads();

        const int arow = waveM * 16 + laneLo;
        #pragma unroll
        for (int ks = 0; ks < KC; ks += 4) {
            v2f a = *(const v2f*)&Xs[arow * XSTR + ks + 2 * laneHi];
            #pragma unroll
            for (int j = 0; j < 3; ++j) {
                v2f b = *(const v2f*)&Ws[((nb + j) * 16 + laneLo) * XSTR + ks + 2 * laneHi];
                acc[j] = __builtin_amdgcn_wmma_f32_16x16x4_f32(
                    false, a, false, b, (short)0, acc[j], false, false);
            }
        }
    }

    __syncthreads();
    // spill logits to LDS: VGPR r holds (M=r, N=lane) / (M=r+8, N=lane-16)
    #pragma unroll
    for (int j = 0; j < 3; ++j) {
        #pragma unroll
        for (int r = 0; r < 8; ++r) {
            int tok = waveM * 16 + r + 8 * laneHi;
            int col = (nb + j) * 16 + laneLo;
            Ls[tok * LSTR + col] = acc[j][r];
        }
    }
    __syncthreads();

    // routing epilogue: one thread per token
    if (tid < 64) {
        const float* Lrow = &Ls[tid * LSTR];
        float g[12];
        float mx = -3.4e38f;
        #pragma unroll
        for (int t = 0; t < 12; ++t) { g[t] = Lrow[84 + t] + gb[t]; mx = fmaxf(mx, g[t]); }
        float s = 0.0f;
        #pragma unroll
        for (int t = 0; t < 12; ++t) { g[t] = expf(g[t] - mx); s += g[t]; }
        const float inv_s = 1.0f / s;

        float* Crow = C + (size_t)(mBase + tid) * NCOLS;
        #pragma unroll
        for (int t = 0; t < 12; ++t) {
            const float wt = g[t] * inv_s;
            float d[7];
            #pragma unroll
            for (int n = 0; n < 7; ++n)
                d[n] = sigmoidf_(Lrow[t * 7 + n] * invT[t * 7 + n] + bo[t * 7 + n]);
            float lp[8];
            #pragma unroll
            for (int l = 0; l < 8; ++l) {
                const int b2 = (l >> 2) & 1, b1 = (l >> 1) & 1, b0 = l & 1;
                float p0 = b2 ? (1.0f - d[0]) : d[0];
                float v1 = d[1 + b2];
                float p1 = b1 ? (1.0f - v1) : v1;
                float v2 = d[3 + 2 * b2 + b1];
                float p2 = b0 ? (1.0f - v2) : v2;
                lp[l] = p0 * p1 * p2 * wt;
            }
            *(float4*)&Crow[t * 8]     = make_float4(lp[0], lp[1], lp[2], lp[3]);
            *(float4*)&Crow[t * 8 + 4] = make_float4(lp[4], lp[5], lp[6], lp[7]);
        }
    }
}

// ---------------------------------------------------------------------------
// Kernel 2: out[h] = C @ leaf_outputs[h]   (M=8192, K=96, N=1024, 3 heads)
// 64x64 tile per WG; full K=96 resident in LDS. A tile fetched by the Tensor
// Data Mover (2D tensor_load_to_lds with pad_enable -> conflict-free stride
// 102); B tile loaded cooperatively and stored transposed.
// ---------------------------------------------------------------------------
__global__ __launch_bounds__(256)
void forest_out_kernel(const float* __restrict__ C,     // [8192][96]
                       const float* __restrict__ leaf,  // [3][96][1024]
                       float* __restrict__ out)         // [3][8192][1024]
{
    __shared__ float smem2[64 * ASTR + 64 * BSTR];
    float* Cs = smem2;               // [64 tokens][96 + pad]  (TDM-written)
    float* Bs = smem2 + 64 * ASTR;   // transposed: [64 cols][96]

    const int tid    = threadIdx.x;
    const int lane   = tid & 31;
    const int w      = tid >> 5;
    const int laneLo = lane & 15;
    const int laneHi = lane >> 4;
    const int nBase  = blockIdx.x * 64;            // column tile within head
    const int mBase  = blockIdx.y * 64;            // token tile
    const int h      = blockIdx.z;                 // head
    const float* leafH = leaf + (size_t)h * NCOLS * D_OUT;

    // --- Tensor Data Mover: async DMA of A tile [64 rows][96 dwords] -> LDS,
    //     2 pad dwords inserted every 32 dwords => padded row stride 102.
    if (w == 0) {
        u32 ldsAddr = (u32)(uintptr_t)(void*)Cs;
        unsigned long long ga =
            (unsigned long long)(uintptr_t)(C + (size_t)mBase * NCOLS);
        u32x4 g0;
        g0.x = 1u;                                        // count=1 (valid user D#)
        g0.y = ldsAddr;                                   // lds_addr (bytes)
        g0.z = (u32)(ga & 0xFFFFFFFFu);                   // global_addr[31:0]
        g0.w = (u32)((ga >> 32) & 0x01FFFFFFu) | (2u << 30); // addr[56:32] | type=2
        i32x8 g1;
        g1[0] = (2 << 16)        // data_size = 4B
              | (1 << 20)        // pad_enable
              | (4 << 22)        // pad_interval: 32 dwords
              | (1 << 25);       // pad_amount: 2 dwords
        g1[1] = (int)((NCOLS & 0xFFFF) << 16);                 // tensor_dim0 lo16
        g1[2] = (int)((NCOLS >> 16) | ((NTOK & 0xFFFF) << 16)); // dim0 hi | dim1 lo
        g1[3] = (int)((NTOK >> 16) | (NCOLS << 16));            // dim1 hi | tile_dim0=96
        g1[4] = 64;                                             // tile_dim1=64, tile_dim2=0
        g1[5] = NCOLS;                                          // tensor_dim0_stride lo32
        g1[6] = 0;
        g1[7] = 0;
        i32x4 z4 = {0, 0, 0, 0};
#if defined(__clang_major__) && (__clang_major__ >= 23)
        i32x8 z8 = {0, 0, 0, 0, 0, 0, 0, 0};
        __builtin_amdgcn_tensor_load_to_lds(g0, g1, z4, z4, z8, 0);
#else
        __builtin_amdgcn_tensor_load_to_lds(g0, g1, z4, z4, 0);
#endif
    }

    // --- B tile: read rows of leafH coalesced along d, store transposed.
    #pragma unroll
    for (int i = 0; i < 24; ++i) {
        int e = tid + i * 256;
        int k = e >> 6, nl = e & 63;
        Bs[nl * BSTR + k] = leafH[(size_t)k * D_OUT + nBase + nl];
    }

    if (w == 0) __builtin_amdgcn_s_wait_tensorcnt(0);
    __syncthreads();

    const int waveM = w & 3;    // 16-token strip
    const int waveN = w >> 2;   // 32-column half
    v8f acc[2] = {};
    const int arow = waveM * 16 + laneLo;

    #pragma unroll
    for (int ks = 0; ks < NCOLS; ks += 4) {
        int ka  = ks + 2 * laneHi;
        int kap = ka + 2 * (ka >> 5);              // TDM pad: +2 dwords per 32
        v2f a = *(const v2f*)&Cs[arow * ASTR + kap];
        #pragma unroll
        for (int jb = 0; jb < 2; ++jb) {
            v2f b = *(const v2f*)&Bs[(waveN * 32 + jb * 16 + laneLo) * BSTR + ks + 2 * laneHi];
            acc[jb] = __builtin_amdgcn_wmma_f32_16x16x4_f32(
                false, a, false, b, (short)0, acc[jb], false, false);
        }
    }

    float* outH = out + (size_t)h * NTOK * D_OUT;
    #pragma unroll
    for (int jb = 0; jb < 2; ++jb) {
        int col = nBase + waveN * 32 + jb * 16 + laneLo;
        #pragma unroll
        for (int r = 0; r < 8; ++r) {
            int row = mBase + waveM * 16 + r + 8 * laneHi;
            outH[(size_t)row * D_OUT + col] = acc[jb][r];
        }
    }
}

// ---------------------------------------------------------------------------
extern "C" void kernel_launch(void* const* d_in, const int* in_sizes, int n_in,
                              void* d_out, int out_size, void* d_ws, size_t ws_size,
                              hipStream_t stream) {
    const float* x    = (const float*)d_in[0];  // [4][2048][1024]
    const float* dw   = (const float*)d_in[1];  // [12][7][1024]
    const float* db   = (const float*)d_in[2];  // [12][7]
    const float* ntl  = (const float*)d_in[3];  // [12][7]
    const float* leaf = (const float*)d_in[4];  // [3][12][8][1024]
    const float* gw   = (const float*)d_in[5];  // [12][1024]
    const float* gb   = (const float*)d_in[6];  // [12]
    float* out = (float*)d_out;                 // [3][4][2048][1024] f32
    float* Cws = (float*)d_ws;                  // [8192][96] f32 = 3 MB scratch

    forest_route_kernel<<<dim3(NTOK / 64), dim3(256), 0, stream>>>(
        x, dw, db, ntl, gw, gb, Cws);
    forest_out_kernel<<<dim3(D_OUT / 64, NTOK / 64, 3), dim3(256), 0, stream>>>(
        Cws, leaf, out);
}